// GAT_Layer_54185307406451
// MI455X (gfx1250) — compile-verified
//
#include <hip/hip_runtime.h>
#include <hip/hip_bf16.h>
#include <math.h>

typedef float v2f __attribute__((ext_vector_type(2)));
typedef float v8f __attribute__((ext_vector_type(8)));

#define DIN 128
#define DOUT 128
#define NEG_SLOPE 0.2f

// ---- order-preserving float<->uint mapping for atomicMax on f32 ----
__device__ __forceinline__ unsigned enc_f32(float x) {
    unsigned u = __float_as_uint(x);
    return (u & 0x80000000u) ? ~u : (u | 0x80000000u);
}
__device__ __forceinline__ float dec_f32(unsigned e) {
    return (e & 0x80000000u) ? __uint_as_float(e ^ 0x80000000u)
                             : __uint_as_float(~e);
}

// ---------------------------------------------------------------------------
// Init: out[b][n][d] = bias[d] (atomic scatter accumulates on top),
// zero per-node stats.
// ---------------------------------------------------------------------------
__global__ void gat_init(float* __restrict__ out, const float* __restrict__ bias,
                         unsigned* __restrict__ amax, float* __restrict__ ssum,
                         float* __restrict__ deg, float* __restrict__ loopsum,
                         int TOT, int BN, int N) {
    int i = blockIdx.x * blockDim.x + threadIdx.x;
    if (i < TOT) out[i] = bias[i & (DOUT - 1)];
    if (i < BN) { amax[i] = 0u; ssum[i] = 0.0f; }
    if (i < N)  { deg[i] = 0.0f; loopsum[i] = 0.0f; }
}

// ---------------------------------------------------------------------------
// Degree + incoming edge_attr sum per dst (for fill_value='mean' self loops)
// ---------------------------------------------------------------------------
__global__ void gat_deg(const int* __restrict__ dst, const float* __restrict__ attr,
                        float* __restrict__ deg, float* __restrict__ loopsum, int E) {
    int e = blockIdx.x * blockDim.x + threadIdx.x;
    if (e >= E) return;
    int d = dst[e];
    atomicAdd(&deg[d], 1.0f);
    atomicAdd(&loopsum[d], attr[e]);
}

__global__ void gat_loopdiv(float* __restrict__ loopsum, const float* __restrict__ deg, int N) {
    int n = blockIdx.x * blockDim.x + threadIdx.x;
    if (n >= N) return;
    loopsum[n] = loopsum[n] / fmaxf(deg[n], 1.0f);
}

// c = dot(We[:,0], att_edge)  (alpha_e = attr * c)
__global__ void gat_cdot(const float* __restrict__ We, const float* __restrict__ att_edge,
                         float* __restrict__ cval) {
    __shared__ float red[128];
    int t = threadIdx.x;
    red[t] = We[t] * att_edge[t];
    __syncthreads();
    for (int s = 64; s > 0; s >>= 1) {
        if (t < s) red[t] += red[t + s];
        __syncthreads();
    }
    if (t == 0) cval[0] = red[0];
}

// ---------------------------------------------------------------------------
// h[b] = x[b] @ W^T via V_WMMA_F32_16X16X4_F32 (wave32, EXEC all-ones).
// One wave per 16x16 output tile; K=128 in 32 steps of 4.
//   A 16x4 f32: lane m=L%16, kh=L/16 -> {A[m][2kh], A[m][2kh+1]}
//   B 4x16 f32: VGPR v -> row K=v+2kh, col n=L%16  (B[k][n] = W[n][k])
//   D 16x16:    VGPR v -> row v+8kh, col L%16
// ---------------------------------------------------------------------------
__global__ void gat_gemm_wmma(const float* __restrict__ x, const float* __restrict__ W,
                              float* __restrict__ h, int N, int rtiles, int tilesTotal) {
    int wave = (blockIdx.x * blockDim.x + threadIdx.x) >> 5;
    int lane = threadIdx.x & 31;
    if (wave >= tilesTotal) return;           // wave-uniform guard
    int c = wave & 7;                          // DOUT/16 = 8 col tiles
    int t = wave >> 3;
    int r = t % rtiles;
    int b = t / rtiles;
    int m  = lane & 15;
    int kh = lane >> 4;

    const float* xrow = x + ((size_t)(b * N + r * 16 + m)) * DIN + 2 * kh;
    const float* wrow = W + (size_t)(c * 16 + m) * DIN + 2 * kh;

    v8f acc = {};
#pragma unroll 4
    for (int k = 0; k < DIN; k += 4) {
        v2f av = *(const v2f*)(xrow + k);
        v2f bv = *(const v2f*)(wrow + k);
        acc = __builtin_amdgcn_wmma_f32_16x16x4_f32(
            /*neg_a=*/false, av, /*neg_b=*/false, bv,
            /*c_mod=*/(short)0, acc, /*reuse_a=*/false, /*reuse_b=*/false);
    }

    float* hout = h + ((size_t)(b * N + r * 16 + 8 * kh)) * DOUT + c * 16 + m;
#pragma unroll
    for (int v = 0; v < 8; ++v)
        hout[(size_t)v * DOUT] = acc[v];
}

// ---------------------------------------------------------------------------
// a_src[b,n] = h[b,n]·att_src ; a_dst likewise. One wave per node row.
// ---------------------------------------------------------------------------
__global__ void gat_adots(const float* __restrict__ h, const float* __restrict__ att_src,
                          const float* __restrict__ att_dst, float* __restrict__ a_src,
                          float* __restrict__ a_dst, int BN) {
    int wave = (blockIdx.x * blockDim.x + threadIdx.x) >> 5;
    int lane = threadIdx.x & 31;
    if (wave >= BN) return;
    const float4 hv = *(const float4*)(h + (size_t)wave * DOUT + lane * 4);
    const float4 as = *(const float4*)(att_src + lane * 4);
    const float4 ad = *(const float4*)(att_dst + lane * 4);
    float ss = hv.x * as.x + hv.y * as.y + hv.z * as.z + hv.w * as.w;
    float sd = hv.x * ad.x + hv.y * ad.y + hv.z * ad.z + hv.w * ad.w;
#pragma unroll
    for (int o = 16; o > 0; o >>= 1) {
        ss += __shfl_xor(ss, o, 32);
        sd += __shfl_xor(sd, o, 32);
    }
    if (lane == 0) { a_src[wave] = ss; a_dst[wave] = sd; }
}

// ---------------------------------------------------------------------------
// Pass 1: raw attention logit + leaky relu; segment max via uint atomicMax
// ---------------------------------------------------------------------------
__global__ void gat_edge1(const int* __restrict__ src, const int* __restrict__ dst,
                          const float* __restrict__ edge_attr, const float* __restrict__ loop_attr,
                          const float* __restrict__ a_src, const float* __restrict__ a_dst,
                          const float* __restrict__ cval, float* __restrict__ attn,
                          unsigned* __restrict__ amax, int N, int E, int E2, int BE2) {
    int i = blockIdx.x * blockDim.x + threadIdx.x;
    if (i >= BE2) return;
    int b = i / E2, e = i % E2;
    int s, d; float at;
    if (e < E) { s = src[e]; d = dst[e]; at = edge_attr[e]; }
    else       { s = d = e - E; at = loop_attr[s]; }
    float a = a_src[b * N + s] + a_dst[b * N + d] + at * cval[0];
    a = (a > 0.0f) ? a : NEG_SLOPE * a;
    attn[i] = a;
    atomicMax(&amax[b * N + d], enc_f32(a));
}

// ---------------------------------------------------------------------------
// Pass 2: ex = exp(a - amax[dst]); segment sum
// ---------------------------------------------------------------------------
__global__ void gat_edge2(const int* __restrict__ dst, const unsigned* __restrict__ amax,
                          float* __restrict__ attn, float* __restrict__ ssum,
                          int N, int E, int E2, int BE2) {
    int i = blockIdx.x * blockDim.x + threadIdx.x;
    if (i >= BE2) return;
    int b = i / E2, e = i % E2;
    int d = (e < E) ? dst[e] : (e - E);
    float ex = __expf(attn[i] - dec_f32(amax[b * N + d]));
    attn[i] = ex;
    atomicAdd(&ssum[b * N + d], ex);
}

// ---------------------------------------------------------------------------
// Pass 3: alpha = ex / (ssum+eps); out[dst] += alpha * h[src].
// One wave per edge; each lane handles 4 features (float4 gather + 4 atomics)
// ---------------------------------------------------------------------------
__global__ void gat_scatter(const int* __restrict__ src, const int* __restrict__ dst,
                            const float* __restrict__ h, const float* __restrict__ ssum,
                            float* __restrict__ attn, float* __restrict__ out,
                            int N, int E, int E2, int BE2) {
    int wave = (blockIdx.x * blockDim.x + threadIdx.x) >> 5;
    int lane = threadIdx.x & 31;
    if (wave >= BE2) return;
    int b = wave / E2, e = wave % E2;
    int s, d;
    if (e < E) { s = src[e]; d = dst[e]; }
    else       { s = d = e - E; }
    float alpha = attn[wave] / (ssum[b * N + d] + 1e-16f);
    if (lane == 0) attn[wave] = alpha;
    const float4 hv = *(const float4*)(h + ((size_t)(b * N + s)) * DOUT + lane * 4);
    float* op = out + ((size_t)(b * N + d)) * DOUT + lane * 4;
    atomicAdd(op + 0, alpha * hv.x);
    atomicAdd(op + 1, alpha * hv.y);
    atomicAdd(op + 2, alpha * hv.z);
    atomicAdd(op + 3, alpha * hv.w);
}

// ---------------------------------------------------------------------------
extern "C" void kernel_launch(void* const* d_in, const int* in_sizes, int n_in,
                              void* d_out, int out_size, void* d_ws, size_t ws_size,
                              hipStream_t stream) {
    const float* x         = (const float*)d_in[0];   // [B,N,128]
    const int*   ei        = (const int*)d_in[1];     // [2,E]
    const float* edge_attr = (const float*)d_in[2];   // [E]
    const float* W         = (const float*)d_in[3];   // [128,128]
    const float* We        = (const float*)d_in[4];   // [128,1]
    const float* att_src   = (const float*)d_in[5];   // [128]
    const float* att_dst   = (const float*)d_in[6];   // [128]
    const float* att_edge  = (const float*)d_in[7];   // [128]
    const float* bias      = (const float*)d_in[8];   // [128]

    const int B  = 8;
    const int E  = in_sizes[2];
    const int N  = in_sizes[0] / (B * DIN);           // 10000 (multiple of 16)
    const int E2 = E + N;
    const int BN = B * N;
    const int BE2 = B * E2;
    const int TOT = BN * DOUT;

    const int* srcI = ei;
    const int* dstI = ei + E;

    float* out  = (float*)d_out;                      // [B,N,DOUT]
    float* attn = out + (size_t)BN * DOUT;            // [B,E2]

    // workspace layout (floats)
    float*    h         = (float*)d_ws;               // BN*DOUT
    float*    a_src     = h + (size_t)BN * DOUT;      // BN
    float*    a_dst     = a_src + BN;                 // BN
    float*    ssum      = a_dst + BN;                 // BN
    unsigned* amax      = (unsigned*)(ssum + BN);     // BN
    float*    deg       = (float*)(amax + BN);        // N
    float*    loop_attr = deg + N;                    // N (sum, then mean in place)
    float*    cval      = loop_attr + N;              // 1

    const int TPB = 256;

    gat_init<<<(TOT + TPB - 1) / TPB, TPB, 0, stream>>>(out, bias, amax, ssum,
                                                        deg, loop_attr, TOT, BN, N);
    gat_deg<<<(E + TPB - 1) / TPB, TPB, 0, stream>>>(dstI, edge_attr, deg, loop_attr, E);
    gat_loopdiv<<<(N + TPB - 1) / TPB, TPB, 0, stream>>>(loop_attr, deg, N);
    gat_cdot<<<1, 128, 0, stream>>>(We, att_edge, cval);

    const int rtiles = N / 16;
    const int tilesTotal = B * rtiles * (DOUT / 16);  // multiple of 8 -> full waves
    gat_gemm_wmma<<<(tilesTotal + 7) / 8, TPB, 0, stream>>>(x, W, h, N, rtiles, tilesTotal);

    gat_adots<<<(BN + 7) / 8, TPB, 0, stream>>>(h, att_src, att_dst, a_src, a_dst, BN);

    gat_edge1<<<(BE2 + TPB - 1) / TPB, TPB, 0, stream>>>(srcI, dstI, edge_attr, loop_attr,
                                                         a_src, a_dst, cval, attn, amax,
                                                         N, E, E2, BE2);
    gat_edge2<<<(BE2 + TPB - 1) / TPB, TPB, 0, stream>>>(dstI, amax, attn, ssum,
                                                         N, E, E2, BE2);
    gat_scatter<<<(BE2 + 7) / 8, TPB, 0, stream>>>(srcI, dstI, h, ssum, attn, out,
                                                   N, E, E2, BE2);
}